// TitleGenerator_64218351010112
// MI455X (gfx1250) — compile-verified
//
#include <hip/hip_runtime.h>
#include <hip/hip_bf16.h>
#include <math.h>

// ---------------------------------------------------------------------------
// TitleGenerator for MI455X (gfx1250, wave32).
//
// Roofline: titles output stream = 174MB -> ~7.5us floor @ 23.3TB/s.
// Compute ~17 GFLOP but serialized over 400 GRU steps, so the design goal is
// minimizing the per-step critical path of a 16-row batch tile:
//   - all GRU weights resident in LDS (~162KB of 320KB/WGP) for 400 steps
//   - every GEMM through V_WMMA_F32_16X16X4_F32 (f32: 400-deep recurrence is
//     precision sensitive, reference is f32)
//   - 8 waves/block (2 per SIMD32), 2 independent WMMA accumulators per tile
//     to break the D->C dependency chain
//   - shuffle-based row max/argmax (no serial 53-element loops), 5 barriers/step
// ---------------------------------------------------------------------------

#define FEAT   4096
#define BATCH  2048
#define NCH    53        // chars / hidden
#define CP     64        // padded char dim (4 x 16-col WMMA tiles)
#define G3     192       // padded 3*H (3 gates x 64)
#define TLEN   400
#define NULLC  52

#define WLD    66        // LDS row stride for [*][64] mats (bank pad, 8B aligned)
#define GLD    196       // LDS row stride for [16][192] gate buffers

typedef __attribute__((ext_vector_type(2))) float v2f;
typedef __attribute__((ext_vector_type(8))) float v8f;

// f32 WMMA 16x16x4: A 16x4 (2 VGPR), B 4x16 (2 VGPR), C/D 16x16 (8 VGPR).
// A: lanes 0-15 -> M=lane, V0=K0,V1=K1 ; lanes 16-31 -> M=lane-16, V0=K2,V1=K3.
// B mirrored (lane covers N=lane%16, hi half covers K+2).
// D: N = lane%16, M = v + 8*(lane/16).
__device__ __forceinline__ v8f wmma_f32(v2f a, v2f b, v8f c) {
    return __builtin_amdgcn_wmma_f32_16x16x4_f32(false, a, false, b, (short)0, c,
                                                 false, false);
}

// ---------------------------------------------------------------------------
// Phase 1: x = lrelu(img_feat @ lin1_w^T + lin1_b)  -> [2048 x 64] padded
// 128 blocks x 16 rows; 4 waves/block, one 16-col tile per wave; K=4096.
// Memory bound (~32MB of img_feat); dual accumulators for WMMA ILP.
// ---------------------------------------------------------------------------
__global__ __launch_bounds__(128)
void lin1_kernel(const float* __restrict__ img, const float* __restrict__ w1,
                 const float* __restrict__ b1, float* __restrict__ xout) {
    const int lane = threadIdx.x & 31;
    const int wave = threadIdx.x >> 5;
    const int r16  = lane & 15;
    const int hi   = lane >> 4;
    const int m0   = blockIdx.x * 16;
    const int n    = wave * 16 + r16;          // B/D column (0..63)

    const float* arow   = img + (size_t)(m0 + r16) * FEAT;
    const float  bscale = (n < NCH) ? 1.0f : 0.0f;          // zero-pad cols 53..63
    const float* brow   = w1 + (size_t)(n < NCH ? n : 0) * FEAT;

    v8f acc0 = {}, acc1 = {};
#pragma unroll 4
    for (int kb = 0; kb < FEAT; kb += 8) {
        if ((kb & 255) == 0) {
            __builtin_prefetch(arow + kb + 512, 0, 0);      // global_prefetch_b8
        }
        v2f a0, b0, a1, b1v;
        a0.x  = arow[kb + 2 * hi + 0];
        a0.y  = arow[kb + 2 * hi + 1];
        b0.x  = brow[kb + 2 * hi + 0] * bscale;
        b0.y  = brow[kb + 2 * hi + 1] * bscale;
        a1.x  = arow[kb + 4 + 2 * hi + 0];
        a1.y  = arow[kb + 4 + 2 * hi + 1];
        b1v.x = brow[kb + 4 + 2 * hi + 0] * bscale;
        b1v.y = brow[kb + 4 + 2 * hi + 1] * bscale;
        acc0 = wmma_f32(a0, b0, acc0);
        acc1 = wmma_f32(a1, b1v, acc1);
    }

    const float bias = (n < NCH) ? b1[n] : 0.0f;
#pragma unroll
    for (int v = 0; v < 8; ++v) {
        const int m = m0 + v + 8 * hi;
        float val = acc0[v] + acc1[v] + bias;
        val = val > 0.0f ? val : 0.01f * val;               // leaky relu
        xout[(size_t)m * CP + n] = (n < NCH) ? val : 0.0f;  // exact-zero padding
    }
}

// ---------------------------------------------------------------------------
// Phase 2: persistent GRU. 128 blocks x 16 batch rows, 8 waves (256 thr).
// ---------------------------------------------------------------------------
__global__ __launch_bounds__(256)
void gru_kernel(const float* __restrict__ xinit,
                const float* __restrict__ wih, const float* __restrict__ whh,
                const float* __restrict__ bih, const float* __restrict__ bhh,
                const float* __restrict__ w2,  const float* __restrict__ b2,
                float* __restrict__ out_titles, int* __restrict__ out_lens) {
    __shared__ float sWih[G3 * WLD];     // w_ih padded [192][64]  (row n, col k)
    __shared__ float sWhh[G3 * WLD];
    __shared__ float sW2 [CP * WLD];     // lin2_w padded [64][64]
    __shared__ float sBih[G3], sBhh[G3], sB2[CP];
    __shared__ float sChars[16 * WLD];   // GRU input  (A matrix)
    __shared__ float sHt   [16 * WLD];   // hidden     (A matrix)
    __shared__ float sLr   [16 * WLD];   // lrelu(ht)  (A matrix for lin2)
    __shared__ float sC    [16 * WLD];   // lin2 output
    __shared__ float sGi   [16 * GLD];   // gi [16][192]
    __shared__ float sGh   [16 * GLD];   // gh [16][192]
    __shared__ float sMax [16];
    __shared__ float sRedV[16 * 8];      // per-(row,chunk) argmax partials
    __shared__ int   sRedI[16 * 8];

    const int tid  = threadIdx.x;
    const int lane = tid & 31;
    const int wave = tid >> 5;
    const int r16  = lane & 15;
    const int hi   = lane >> 4;
    const int m0   = blockIdx.x * 16;

    // ---- stage weights (zero-padded) into LDS, once ----
    for (int i = tid; i < G3 * CP; i += 256) {
        const int nn = i / CP, k = i % CP;
        const int g = nn >> 6, j = nn & 63;
        float vih = 0.0f, vhh = 0.0f;
        if (j < NCH && k < NCH) {
            vih = wih[(g * NCH + j) * NCH + k];
            vhh = whh[(g * NCH + j) * NCH + k];
        }
        sWih[nn * WLD + k] = vih;
        sWhh[nn * WLD + k] = vhh;
    }
    for (int i = tid; i < CP * CP; i += 256) {
        const int nn = i / CP, k = i % CP;
        sW2[nn * WLD + k] = (nn < NCH && k < NCH) ? w2[nn * NCH + k] : 0.0f;
    }
    for (int i = tid; i < G3; i += 256) {
        const int g = i >> 6, j = i & 63;
        sBih[i] = (j < NCH) ? bih[g * NCH + j] : 0.0f;
        sBhh[i] = (j < NCH) ? bhh[g * NCH + j] : 0.0f;
    }
    for (int i = tid; i < CP; i += 256) sB2[i] = (i < NCH) ? b2[i] : 0.0f;
    for (int i = tid; i < 16 * WLD; i += 256) {
        sHt[i] = 0.0f; sLr[i] = 0.0f; sChars[i] = 0.0f; sC[i] = 0.0f;
    }
    __syncthreads();
    for (int i = tid; i < 16 * CP; i += 256) {
        const int b = i / CP, j = i % CP;
        sChars[b * WLD + j] = xinit[(size_t)(m0 + b) * CP + j];  // padded 0
    }
    __syncthreads();

    int firstNull = -1;   // tracked by tid<16 (one per batch row)

    for (int t = 0; t < TLEN; ++t) {
        // ---- GEMM1: gi = chars@wih^T, gh = ht@whh^T (24 tile jobs / 8 waves)
        for (int job = wave; job < 24; job += 8) {
            const int    mat = job / 12;              // 0 -> gi, 1 -> gh
            const int    n0  = (job % 12) * 16;
            const float* A   = mat ? sHt  : sChars;
            const float* Bm  = mat ? sWhh : sWih;
            float*       D   = mat ? sGh  : sGi;
            v8f acc0 = {}, acc1 = {};
#pragma unroll
            for (int ks = 0; ks < 16; ks += 2) {      // K = 64, dual accum
                const int kb0 = ks * 4 + 2 * hi;
                const int kb1 = kb0 + 4;
                v2f a0, b0, a1, b1v;
                a0.x  = A[r16 * WLD + kb0];       a0.y  = A[r16 * WLD + kb0 + 1];
                b0.x  = Bm[(n0 + r16) * WLD + kb0]; b0.y = Bm[(n0 + r16) * WLD + kb0 + 1];
                a1.x  = A[r16 * WLD + kb1];       a1.y  = A[r16 * WLD + kb1 + 1];
                b1v.x = Bm[(n0 + r16) * WLD + kb1]; b1v.y = Bm[(n0 + r16) * WLD + kb1 + 1];
                acc0 = wmma_f32(a0, b0, acc0);
                acc1 = wmma_f32(a1, b1v, acc1);
            }
#pragma unroll
            for (int v = 0; v < 8; ++v)
                D[(v + 8 * hi) * GLD + n0 + r16] = acc0[v] + acc1[v];
        }
        __syncthreads();

        // ---- elementwise gates: r, z, n, ht_new, lrelu(ht_new)
        for (int i = tid; i < 16 * NCH; i += 256) {
            const int b = i / NCH, j = i % NCH;
            const float gr  = sGi[b * GLD + j]        + sBih[j] +
                              sGh[b * GLD + j]        + sBhh[j];
            const float gz  = sGi[b * GLD + 64 + j]   + sBih[64 + j] +
                              sGh[b * GLD + 64 + j]   + sBhh[64 + j];
            const float gin = sGi[b * GLD + 128 + j]  + sBih[128 + j];
            const float ghn = sGh[b * GLD + 128 + j]  + sBhh[128 + j];
            const float r = 1.0f / (1.0f + __expf(-gr));
            const float z = 1.0f / (1.0f + __expf(-gz));
            const float n = tanhf(gin + r * ghn);
            const float h = (1.0f - z) * n + z * sHt[b * WLD + j];
            sHt[b * WLD + j] = h;
            sLr[b * WLD + j] = h > 0.0f ? h : 0.01f * h;
        }
        __syncthreads();

        // ---- GEMM2: c = lrelu(ht) @ lin2_w^T + b2 (waves 0..3, 1 tile each)
        if (wave < 4) {
            const int n0 = wave * 16;
            v8f acc0 = {}, acc1 = {};
#pragma unroll
            for (int ks = 0; ks < 16; ks += 2) {
                const int kb0 = ks * 4 + 2 * hi;
                const int kb1 = kb0 + 4;
                v2f a0, b0, a1, b1v;
                a0.x  = sLr[r16 * WLD + kb0];        a0.y  = sLr[r16 * WLD + kb0 + 1];
                b0.x  = sW2[(n0 + r16) * WLD + kb0]; b0.y  = sW2[(n0 + r16) * WLD + kb0 + 1];
                a1.x  = sLr[r16 * WLD + kb1];        a1.y  = sLr[r16 * WLD + kb1 + 1];
                b1v.x = sW2[(n0 + r16) * WLD + kb1]; b1v.y = sW2[(n0 + r16) * WLD + kb1 + 1];
                acc0 = wmma_f32(a0, b0, acc0);
                acc1 = wmma_f32(a1, b1v, acc1);
            }
#pragma unroll
            for (int v = 0; v < 8; ++v)
                sC[(v + 8 * hi) * WLD + n0 + r16] =
                    acc0[v] + acc1[v] + sB2[n0 + r16];
        }
        __syncthreads();

        // ---- row max (shuffle butterfly) + argmax-of-threshold partials
        if (tid < 128) {
            const int row = tid >> 3, sub = tid & 7;
            float m = -3.4e38f;
            for (int j = sub; j < NCH; j += 8) m = fmaxf(m, sC[row * WLD + j]);
            m = fmaxf(m, __shfl_xor(m, 1, 8));     // width-8 butterfly, wave32
            m = fmaxf(m, __shfl_xor(m, 2, 8));
            m = fmaxf(m, __shfl_xor(m, 4, 8));
            if (sub == 0) sMax[row] = m;
            // partial argmax of thresholded values within this lane's chunk
            float best = 0.0f; int bi = sub;
            for (int j = sub; j < NCH; j += 8) {
                const float cn = sC[row * WLD + j] / m;
                const float tv = cn > 1.0f ? cn : 0.0f;
                if (tv > best) { best = tv; bi = j; }
            }
            sRedV[row * 8 + sub] = best;
            sRedI[row * 8 + sub] = bi;
        }
        __syncthreads();

        // ---- threshold, stream titles, feed back as next chars; combine argmax
        for (int i = tid; i < 16 * NCH; i += 256) {
            const int b = i / NCH, j = i % NCH;
            const float cn = sC[b * WLD + j] / sMax[b];
            const float tv = cn > 1.0f ? cn : 0.0f;
            sChars[b * WLD + j] = tv;              // cols 53..63 stay 0
            out_titles[((size_t)(m0 + b) * TLEN + t) * NCH + j] = tv;
        }
        if (tid < 16) {                            // first-max tie-break (jnp.argmax)
            float best = 0.0f; int bi = 0;
            for (int s = 0; s < 8; ++s) {
                const float v = sRedV[tid * 8 + s];
                const int   i = sRedI[tid * 8 + s];
                if (v > best || (v == best && v > 0.0f && i < bi)) { best = v; bi = i; }
            }
            if (firstNull < 0 && bi == NULLC) firstNull = t;
        }
        __syncthreads();   // also protects sChars before next step's GEMM1
    }

    if (tid < 16)
        out_lens[m0 + tid] = (firstNull >= 0) ? firstNull + 1 : 0;
}

// ---------------------------------------------------------------------------
extern "C" void kernel_launch(void* const* d_in, const int* in_sizes, int n_in,
                              void* d_out, int out_size, void* d_ws, size_t ws_size,
                              hipStream_t stream) {
    const float* img = (const float*)d_in[0];
    const float* w1  = (const float*)d_in[1];
    const float* b1  = (const float*)d_in[2];
    const float* wih = (const float*)d_in[3];
    const float* whh = (const float*)d_in[4];
    const float* bih = (const float*)d_in[5];
    const float* bhh = (const float*)d_in[6];
    const float* w2  = (const float*)d_in[7];
    const float* b2  = (const float*)d_in[8];

    float* out = (float*)d_out;
    float* xws = (float*)d_ws;                       // [BATCH][CP] padded x
    const size_t titles_elems = (size_t)BATCH * TLEN * NCH;
    int* lens = (int*)(out + titles_elems);          // int32 lens after titles

    lin1_kernel<<<BATCH / 16, 128, 0, stream>>>(img, w1, b1, xws);
    gru_kernel <<<BATCH / 16, 256, 0, stream>>>(xws, wih, whh, bih, bhh, w2, b2,
                                                out, lens);
}